// Model_31791347925195
// MI455X (gfx1250) — compile-verified
//
#include <hip/hip_runtime.h>
#include <hip/hip_bf16.h>
#include <math.h>

typedef __attribute__((ext_vector_type(2))) float v2f;
typedef __attribute__((ext_vector_type(8))) float v8f;

#define IMG   256
#define NPIX  (IMG*IMG)
#define NFACE 512
#define NTILE (NFACE/16)

// ws layout (floats):
//   [0      .. 12287] aFrag : 2 renders * 3 quantities * 32 tiles * 32 lanes * 2 floats
//   [12288  .. 24575] texTanh : 512 faces * 24
//   [24576  .. 25599] block partial sums (1024)
#define WS_AFRAG 0
#define WS_TEX   12288
#define WS_PART  24576

struct CamParams {
    float R[2][9];
    float eye[2][3];
};

// ---------------------------------------------------------------------------
// tanh(textures) precompute: 512*2*2*2*3 = 12288 elements
// ---------------------------------------------------------------------------
__global__ void tanh_kernel(const float* __restrict__ tex_in,
                            float* __restrict__ tex_out, int n) {
    int i = blockIdx.x * blockDim.x + threadIdx.x;
    if (i < n) tex_out[i] = tanhf(tex_in[i]);
}

// ---------------------------------------------------------------------------
// Per-(render, face) affine coefficients for w0, w1, depth as functions of
// (px, py, 1).  Written directly in V_WMMA_F32_16X16X4_F32 A-fragment layout:
//   lanes 0-15  hold K=0,1 -> (A, B)
//   lanes 16-31 hold K=2,3 -> (C, 0)
// ---------------------------------------------------------------------------
__global__ void coeff_kernel(const float* __restrict__ verts,
                             const int* __restrict__ faces,
                             float* __restrict__ afrag,
                             CamParams cam) {
    int t = blockIdx.x * blockDim.x + threadIdx.x;
    if (t >= 2 * NFACE) return;
    int rr = t / NFACE;
    int f  = t % NFACE;
    const float* R = cam.R[rr];
    const float* e = cam.eye[rr];

    float X[3], Y[3], Z[3];
    for (int k = 0; k < 3; ++k) {
        int vi = faces[f * 3 + k];
        float dx = verts[vi * 3 + 0] - e[0];
        float dy = verts[vi * 3 + 1] - e[1];
        float dz = verts[vi * 3 + 2] - e[2];
        X[k] = R[0] * dx + R[1] * dy + R[2] * dz;
        Y[k] = R[3] * dx + R[4] * dy + R[5] * dz;
        Z[k] = R[6] * dx + R[7] * dy + R[8] * dz;
    }

    float den = (Y[1] - Y[2]) * (X[0] - X[2]) + (X[2] - X[1]) * (Y[0] - Y[2]);
    if (fabsf(den) < 1e-9f) den = 1e-9f;

    float A0 = (Y[1] - Y[2]) / den, B0 = (X[2] - X[1]) / den;
    float C0 = -(A0 * X[2] + B0 * Y[2]);
    float A1 = (Y[2] - Y[0]) / den, B1 = (X[0] - X[2]) / den;
    float C1 = -(A1 * X[2] + B1 * Y[2]);
    float dz0 = Z[0] - Z[2], dz1 = Z[1] - Z[2];
    float AD = A0 * dz0 + A1 * dz1;
    float BD = B0 * dz0 + B1 * dz1;
    float CD = C0 * dz0 + C1 * dz1 + Z[2];

    float Aq[3] = {A0, A1, AD};
    float Bq[3] = {B0, B1, BD};
    float Cq[3] = {C0, C1, CD};

    int tile = f >> 4, i = f & 15;
    for (int q = 0; q < 3; ++q) {
        int base = ((rr * 3 + q) * NTILE + tile) * 64;   // 32 lanes * 2 floats
        afrag[base + i * 2 + 0]        = Aq[q];
        afrag[base + i * 2 + 1]        = Bq[q];
        afrag[base + (16 + i) * 2 + 0] = Cq[q];
        afrag[base + (16 + i) * 2 + 1] = 0.0f;
    }
}

// ---------------------------------------------------------------------------
// Rasterizer: one wave = 16 pixels.  Loop over 32 face tiles:
// 3x V_WMMA_F32_16X16X4_F32 (A=face coeffs, B=pixel coords) producing
// w0/w1/depth for a 16-face x 16-pixel tile, then branchless z-buffer argmin.
// ---------------------------------------------------------------------------
__global__ __launch_bounds__(256) void raster_kernel(
        const float* __restrict__ afrag,
        const float* __restrict__ texTanh,
        const float* __restrict__ ref0,
        const float* __restrict__ ref1,
        float* __restrict__ partials) {
    const int lane = threadIdx.x & 31;
    const int wave = threadIdx.x >> 5;
    const int g    = blockIdx.x * 8 + wave;   // 0..8191 pixel-groups
    const int rr   = g >> 12;                 // render index (0/1)
    const int pixbase = (g & 4095) << 4;      // first of 16 pixels
    const int n = lane & 15;
    const int p = pixbase + n;
    const int xi = p & 255, yi = p >> 8;

    const float cx =  (2.0f * (float)xi + 1.0f) * (1.0f / 256.0f) - 1.0f;
    const float cy = -((2.0f * (float)yi + 1.0f) * (1.0f / 256.0f) - 1.0f);
    const bool hi = lane >= 16;

    // B fragment (4x16 pixel matrix): lanes 0-15 rows K=0,1 = (px,py);
    // lanes 16-31 rows K=2,3 = (1,0)
    v2f bf;
    bf.x = hi ? 1.0f : cx;
    bf.y = hi ? 0.0f : cy;

    const float* a0p = afrag + (rr * 3 + 0) * (NTILE * 64);
    const float* a1p = afrag + (rr * 3 + 1) * (NTILE * 64);
    const float* a2p = afrag + (rr * 3 + 2) * (NTILE * 64);

    const float INF = __builtin_inff();
    float bz   = INF;
    int   bfid = 0x7fffffff;
    float bw0  = 0.0f, bw1 = 0.0f;

    for (int t = 0; t < NTILE; ++t) {
        v2f a0, a1, a2;
        a0.x = a0p[t * 64 + lane * 2 + 0]; a0.y = a0p[t * 64 + lane * 2 + 1];
        a1.x = a1p[t * 64 + lane * 2 + 0]; a1.y = a1p[t * 64 + lane * 2 + 1];
        a2.x = a2p[t * 64 + lane * 2 + 0]; a2.y = a2p[t * 64 + lane * 2 + 1];

        v8f z8 = {};
        v8f W0 = __builtin_amdgcn_wmma_f32_16x16x4_f32(false, a0, false, bf,
                                                       (short)0, z8, false, false);
        v8f W1 = __builtin_amdgcn_wmma_f32_16x16x4_f32(false, a1, false, bf,
                                                       (short)0, z8, false, false);
        v8f DD = __builtin_amdgcn_wmma_f32_16x16x4_f32(false, a2, false, bf,
                                                       (short)0, z8, false, false);

        // D layout: VGPR r holds face row M = r (+8 for lanes 16-31), pixel N = lane%16
        const int fbase = t * 16 + (hi ? 8 : 0);
#pragma unroll
        for (int r = 0; r < 8; ++r) {
            float w0 = W0[r], w1 = W1[r];
            // inside <=> min(w0, w1, 1-w0-w1) >= 0
            float m = fminf(fminf(w0, w1), 1.0f - w0 - w1);
            float z = (m >= 0.0f) ? DD[r] : INF;
            // Branchless first-min argmin (faces visited in ascending order,
            // strict < keeps the earliest).  Lowers to v_cmp + v_cndmask,
            // no EXEC manipulation between WMMAs.
            bool better = z < bz;
            int  nf = fbase + r;
            bz   = better ? z  : bz;
            bfid = better ? nf : bfid;
            bw0  = better ? w0 : bw0;
            bw1  = better ? w1 : bw1;
        }
    }

    // Combine lane halves: lane and lane^16 hold the same pixel, disjoint faces.
    {
        float oz = __shfl_xor(bz, 16, 32);
        int   of = __shfl_xor(bfid, 16, 32);
        float o0 = __shfl_xor(bw0, 16, 32);
        float o1 = __shfl_xor(bw1, 16, 32);
        bool take = (oz < bz) || ((oz == bz) && (of < bfid));
        bz   = take ? oz : bz;
        bfid = take ? of : bfid;
        bw0  = take ? o0 : bw0;
        bw1  = take ? o1 : bw1;
    }

    float loss = 0.0f;
    if (!hi) {
        float col0 = 0.0f, col1 = 0.0f, col2 = 0.0f;
        const bool hit = bz < INF;
        if (hit) {
            float u = fminf(fmaxf(bw0, 0.0f), 1.0f);
            float v = fminf(fmaxf(bw1, 0.0f), 1.0f);
            float s = fminf(fmaxf(1.0f - bw0 - bw1, 0.0f), 1.0f);
            const float* tex = texTanh + bfid * 24;
            float wa[2] = {1.0f - u, u};
            float wb[2] = {1.0f - v, v};
            float wc[2] = {1.0f - s, s};
#pragma unroll
            for (int a = 0; a < 2; ++a)
#pragma unroll
                for (int b = 0; b < 2; ++b)
#pragma unroll
                    for (int c = 0; c < 2; ++c) {
                        float w = wa[a] * wb[b] * wc[c];
                        int o = ((a * 2 + b) * 2 + c) * 3;
                        col0 += tex[o + 0] * w;
                        col1 += tex[o + 1] * w;
                        col2 += tex[o + 2] * w;
                    }
        }
        const float* ref = rr ? ref1 : ref0;
        float d0 = col0 - ref[0 * NPIX + p];
        float d1 = col1 - ref[1 * NPIX + p];
        float d2 = col2 - ref[2 * NPIX + p];
        loss = 0.5f * (d0 * d0 + d1 * d1 + d2 * d2);   // (L1+L2)/2 folded in
    }

    __shared__ float red[256];
    red[threadIdx.x] = loss;
    __syncthreads();
    for (int s = 128; s > 0; s >>= 1) {
        if (threadIdx.x < s) red[threadIdx.x] += red[threadIdx.x + s];
        __syncthreads();
    }
    if (threadIdx.x == 0) partials[blockIdx.x] = red[0];
}

// ---------------------------------------------------------------------------
// Deterministic fixed-order final reduction of 1024 partials.
// ---------------------------------------------------------------------------
__global__ void reduce_kernel(const float* __restrict__ partials,
                              float* __restrict__ out) {
    __shared__ float red[256];
    int i = threadIdx.x;
    float s = partials[i] + partials[i + 256] + partials[i + 512] + partials[i + 768];
    red[i] = s;
    __syncthreads();
    for (int k = 128; k > 0; k >>= 1) {
        if (i < k) red[i] += red[i + k];
        __syncthreads();
    }
    if (i == 0) out[0] = red[0];
}

// ---------------------------------------------------------------------------
// Host: replicate _get_points_from_angles + _look_at in f32.
// ---------------------------------------------------------------------------
static void compute_cam(float elev_deg, float azim_deg, float* R, float* eye) {
    const float dist = 2.732f;
    const float d2r = 3.14159265358979323846f / 180.0f;
    float e = elev_deg * d2r, a = azim_deg * d2r;
    eye[0] = dist * cosf(e) * sinf(a);
    eye[1] = dist * sinf(e);
    eye[2] = -dist * cosf(e) * cosf(a);
    // z = normalize(-eye)
    float zx = -eye[0], zy = -eye[1], zz = -eye[2];
    float zn = sqrtf(zx * zx + zy * zy + zz * zz) + 1e-12f;
    zx /= zn; zy /= zn; zz /= zn;
    // x = normalize(cross(up, z)), up = (0,1,0) -> (zz, 0, -zx)
    float xx = zz, xy = 0.0f, xz = -zx;
    float xn = sqrtf(xx * xx + xy * xy + xz * xz) + 1e-12f;
    xx /= xn; xy /= xn; xz /= xn;
    // y = cross(z, x)
    float yx = zy * xz - zz * xy;
    float yy = zz * xx - zx * xz;
    float yz = zx * xy - zy * xx;
    R[0] = xx; R[1] = xy; R[2] = xz;
    R[3] = yx; R[4] = yy; R[5] = yz;
    R[6] = zx; R[7] = zy; R[8] = zz;
}

extern "C" void kernel_launch(void* const* d_in, const int* in_sizes, int n_in,
                              void* d_out, int out_size, void* d_ws, size_t ws_size,
                              hipStream_t stream) {
    const float* verts   = (const float*)d_in[0];   // (258,3)
    const float* texture = (const float*)d_in[1];   // (512,2,2,2,3)
    const float* ref0    = (const float*)d_in[2];   // (3,256,256)
    const float* ref1    = (const float*)d_in[3];   // (3,256,256)
    const int*   faces   = (const int*)d_in[4];     // (512,3)
    float* out = (float*)d_out;
    float* ws  = (float*)d_ws;

    float* afrag    = ws + WS_AFRAG;
    float* texTanh  = ws + WS_TEX;
    float* partials = ws + WS_PART;

    CamParams cam;
    compute_cam(180.0f, 0.0f, cam.R[0], cam.eye[0]);
    compute_cam(0.0f,   0.0f, cam.R[1], cam.eye[1]);

    tanh_kernel<<<48, 256, 0, stream>>>(texture, texTanh, NFACE * 24);
    coeff_kernel<<<4, 256, 0, stream>>>(verts, faces, afrag, cam);
    raster_kernel<<<1024, 256, 0, stream>>>(afrag, texTanh, ref0, ref1, partials);
    reduce_kernel<<<1, 256, 0, stream>>>(partials, out);
    (void)in_sizes; (void)n_in; (void)out_size; (void)ws_size;
}